// GIN_46909632807735
// MI455X (gfx1250) — compile-verified
//
#include <hip/hip_runtime.h>
#include <math.h>

typedef __attribute__((ext_vector_type(2))) float v2f;
typedef __attribute__((ext_vector_type(8))) float v8f;

#define D_IN 128
#define D_H  256

// ---------------------------------------------------------------------------
// Scatter-add: agg[dst[e]][:] += x[src[e]][:]   (float4 chunks, one chunk/thread)
// F = 4 << shift features per row (shift=5 -> 128, shift=6 -> 256).
// Consecutive lanes cover consecutive float4 chunks of one edge row -> coalesced.
// ---------------------------------------------------------------------------
__global__ void __launch_bounds__(256)
gin_scatter_add(const float* __restrict__ x,
                const int* __restrict__ src,
                const int* __restrict__ dst,
                float* __restrict__ agg,
                int E, int shift) {
    long long total = (long long)E << shift;
    long long i = (long long)blockIdx.x * blockDim.x + threadIdx.x;
    if (i >= total) return;
    int e = (int)(i >> shift);
    int c = (int)(i & ((1 << shift) - 1));
    int s = src[e];
    int d = dst[e];
    const float4* xs = (const float4*)x;
    float4 v = xs[((long long)s << shift) + c];
    float* o = agg + ((long long)d << (shift + 2)) + ((long long)c << 2);
    atomicAdd(o + 0, v.x);
    atomicAdd(o + 1, v.y);
    atomicAdd(o + 2, v.z);
    atomicAdd(o + 3, v.w);
}

// ---------------------------------------------------------------------------
// OUT[M x N] = act( (in0 [+ in1]) [M x K] @ W [K x N] + bias ), fp32 WMMA.
//
// Block = 8 waves, all sharing one 16-column W panel (nt = blockIdx.y) staged
// in LDS; each wave computes one 16x16 output tile (mt = blockIdx.x*8+wave)
// via K/4 x v_wmma_f32_16x16x4_f32.
//
// Per-lane ISA layout (CDNA5 WMMA, wave32):
//   A 16x4 : lanes 0-15 M=lane, VGPR0=K0 VGPR1=K1; lanes 16-31 K2/K3
//   B 4x16 : lane = n-col; VGPR0 = rows K0(lo)/K2(hi), VGPR1 = K1(lo)/K3(hi)
//   C 16x16: VGPR v -> row v (lanes 0-15) / v+8 (lanes 16-31), col = lane&15
//
// Requires M%16==0, N%16==0, K%4==0, K<=256.
// ---------------------------------------------------------------------------
template <bool HAS_ADD>
__global__ void __launch_bounds__(256)
gin_gemm_wmma_f32(const float* __restrict__ in0,
                  const float* __restrict__ in1,   // additive A input (x+agg fuse)
                  const float* __restrict__ W,     // K x N row-major
                  const float* __restrict__ bias,  // N
                  float* __restrict__ out,         // M x N
                  int M, int K, int N, int doRelu) {
    __shared__ float Bs[D_H * 16];                 // W[:, nt*16 .. +16), K x 16, <=16KB

    const int lane = threadIdx.x;                  // 0..31
    const int wave = threadIdx.y;                  // 0..7
    const int nt   = blockIdx.y;
    const int Mtiles = M >> 4;

    int mt = blockIdx.x * 8 + wave;
    const bool valid = (mt < Mtiles);
    if (!valid) mt = Mtiles - 1;                   // clamp: stay alive for barrier,
                                                   // keep EXEC all-1 for WMMA

    // ---- stage B panel into LDS (16 consecutive lanes read 64B of one W row)
    {
        const int t = wave * 32 + lane;            // 0..255
        const float* wp = W + (size_t)nt * 16;
        for (int idx = t; idx < K * 16; idx += 256) {
            const int kk = idx >> 4;
            const int cc = idx & 15;
            Bs[idx] = wp[(size_t)kk * N + cc];
        }
    }
    __syncthreads();

    const int half = lane >> 4;                    // 0: lanes 0-15, 1: lanes 16-31
    const int mr   = lane & 15;
    const int ncol = nt * 16 + mr;

    // A fragment per lane = two consecutive floats -> one global_load_b64
    const float* a0 = in0 + (size_t)(mt * 16 + mr) * K + half * 2;
    const float* a1 = HAS_ADD ? (in1 + (size_t)(mt * 16 + mr) * K + half * 2) : nullptr;

    v8f c = {};
#pragma unroll 4
    for (int k = 0; k < K; k += 4) {
        float2 av = *(const float2*)(a0 + k);      // 8B-aligned (k0 even)
        if (HAS_ADD) {
            const float2 a2 = *(const float2*)(a1 + k);
            av.x += a2.x;
            av.y += a2.y;
        }
        const int k0 = k + half * 2;
        v2f a, b;
        a.x = av.x;
        a.y = av.y;
        b.x = Bs[k0 * 16 + mr];
        b.y = Bs[(k0 + 1) * 16 + mr];
        // 8 args: (neg_a, A, neg_b, B, c_mod, C, reuse_a, reuse_b)
        c = __builtin_amdgcn_wmma_f32_16x16x4_f32(false, a, false, b,
                                                  (short)0, c, false, false);
    }

    if (valid) {
        const float bv = bias[ncol];
#pragma unroll
        for (int v = 0; v < 8; ++v) {
            float r = c[v] + bv;
            if (doRelu) r = fmaxf(r, 0.0f);
            out[(size_t)(mt * 16 + v + half * 8) * N + ncol] = r;
        }
    }
}

// ---------------------------------------------------------------------------
// Head: logits = h[node] @ w4(256x2) + b4; write log_softmax. One wave/node.
// ---------------------------------------------------------------------------
__global__ void __launch_bounds__(256)
gin_head_logsoftmax(const float* __restrict__ h,
                    const float* __restrict__ w4,   // 256 x 2 row-major
                    const float* __restrict__ b4,   // 2
                    float* __restrict__ out,        // M x 2
                    int M) {
    const int lane = threadIdx.x;
    const int node = blockIdx.x * blockDim.y + threadIdx.y;
    if (node >= M) return;
    const float* hr = h + (long long)node * D_H;
    float s0 = 0.0f, s1 = 0.0f;
    for (int k = lane; k < D_H; k += 32) {
        const float v = hr[k];
        s0 += v * w4[k * 2 + 0];
        s1 += v * w4[k * 2 + 1];
    }
    for (int off = 16; off > 0; off >>= 1) {
        s0 += __shfl_down(s0, off, 32);
        s1 += __shfl_down(s1, off, 32);
    }
    if (lane == 0) {
        const float v0 = s0 + b4[0];
        const float v1 = s1 + b4[1];
        const float m  = fmaxf(v0, v1);
        const float lse = m + logf(expf(v0 - m) + expf(v1 - m));
        out[(long long)node * 2 + 0] = v0 - lse;
        out[(long long)node * 2 + 1] = v1 - lse;
    }
}

// ---------------------------------------------------------------------------
// Driver
// ---------------------------------------------------------------------------
extern "C" void kernel_launch(void* const* d_in, const int* in_sizes, int n_in,
                              void* d_out, int out_size, void* d_ws, size_t ws_size,
                              hipStream_t stream) {
    const float* x  = (const float*)d_in[0];
    const int*   ei = (const int*)d_in[1];
    const float* w1 = (const float*)d_in[2];
    const float* b1 = (const float*)d_in[3];
    const float* w2 = (const float*)d_in[4];
    const float* b2 = (const float*)d_in[5];
    const float* w3 = (const float*)d_in[6];
    const float* b3 = (const float*)d_in[7];
    const float* w4 = (const float*)d_in[8];
    const float* b4 = (const float*)d_in[9];
    float* out = (float*)d_out;

    const int Mn = in_sizes[0] / D_IN;   // 50000 nodes
    const int E  = in_sizes[1] / 2;      // 600000 edges
    const int* src = ei;
    const int* dst = ei + E;

    // Workspace: three Mn x 256 fp32 buffers (agg A, temp B, hidden C).
    float* A = (float*)d_ws;
    float* B = A + (size_t)Mn * D_H;
    float* C = B + (size_t)Mn * D_H;

    const dim3 blk256(256, 1, 1);
    const dim3 gblk(32, 8, 1);                       // 8 waves per block
    const int  Mtiles = Mn / 16;
    const dim3 ggrid((Mtiles + 7) / 8, D_H / 16, 1); // x: m-tiles, y: 16-col W panel

    // ---------------- Layer 1 ----------------
    hipMemsetAsync(A, 0, (size_t)Mn * D_IN * sizeof(float), stream);
    {
        const long long total = (long long)E * (D_IN / 4);
        const int blocks = (int)((total + 255) / 256);
        gin_scatter_add<<<blocks, blk256, 0, stream>>>(x, src, dst, A, E, 5);
    }
    // B = relu((x + agg) @ w1 + b1)
    gin_gemm_wmma_f32<true><<<ggrid, gblk, 0, stream>>>(x, A, w1, b1, B, Mn, D_IN, D_H, 1);
    // C = relu(B @ w2 + b2)   (MLP output + inter-layer F.relu fused)
    gin_gemm_wmma_f32<false><<<ggrid, gblk, 0, stream>>>(B, nullptr, w2, b2, C, Mn, D_H, D_H, 1);

    // ---------------- Layer 2 ----------------
    hipMemsetAsync(A, 0, (size_t)Mn * D_H * sizeof(float), stream);
    {
        const long long total = (long long)E * (D_H / 4);
        const int blocks = (int)((total + 255) / 256);
        gin_scatter_add<<<blocks, blk256, 0, stream>>>(C, src, dst, A, E, 6);
    }
    // B = relu((C + agg) @ w3 + b3)
    gin_gemm_wmma_f32<true><<<ggrid, gblk, 0, stream>>>(C, A, w3, b3, B, Mn, D_H, D_H, 1);

    // ---------------- Head + log_softmax ----------------
    {
        const int blocks = (Mn + 7) / 8;
        gin_head_logsoftmax<<<blocks, gblk, 0, stream>>>(B, w4, b4, out, Mn);
    }
}